// LSM_49048526520353
// MI455X (gfx1250) — compile-verified
//
#include <hip/hip_runtime.h>
#include <hip/hip_bf16.h>
#include <math.h>

// Problem constants (match reference)
#define NS    4096          // S: sampled nodes
#define NE    (1 << 20)     // E: edges
#define EPSF  1e-6f
#define TILES_PER_DIM 256   // S / 16
#define NUM_TILES (TILES_PER_DIM * TILES_PER_DIM)   // 65536
#define PAIR_BLOCKS (NUM_TILES / 8)                 // 8 waves (tiles) per block
#define EDGE_BLOCKS 1024

// Workspace layout (floats)
#define WS_ZS   0           // [S*8]  gathered sampled Z rows
#define WS_BS   32768       // [S]    gathered beta
#define WS_NS   36864       // [S]    ||Z||^2 per sample
#define WS_SS   40960       // [S]    sum_d Z_d per sample
#define WS_P1   45056       // [PAIR_BLOCKS] pair-term partials (already * -0.5)
#define WS_P2   53248       // [EDGE_BLOCKS] edge-term partials

typedef float v2f __attribute__((ext_vector_type(2)));
typedef float v8f __attribute__((ext_vector_type(8)));

// ---------------------------------------------------------------------------
// Kernel 1: gather sampled rows, compute per-sample norm^2 and row-sum.
// ---------------------------------------------------------------------------
__global__ __launch_bounds__(256) void lsm_gather_kernel(
    const float* __restrict__ beta, const float* __restrict__ Z,
    const int* __restrict__ sidx, float* __restrict__ ws) {
  int s = blockIdx.x * blockDim.x + threadIdx.x;
  if (s >= NS) return;
  int idx = sidx[s];
  const float* zr = Z + (size_t)idx * 8;
  float* Zs = ws + WS_ZS;
  float nrm = 0.f, sm = 0.f;
#pragma unroll
  for (int d = 0; d < 8; ++d) {
    float v = zr[d];
    Zs[s * 8 + d] = v;
    nrm = fmaf(v, v, nrm);
    sm += v;
  }
  ws[WS_BS + s] = beta[idx];
  ws[WS_NS + s] = nrm;
  ws[WS_SS + s] = sm;
}

// ---------------------------------------------------------------------------
// Kernel 2: pair term. One 16x16 Gram tile per wave via V_WMMA_F32_16X16X4_F32
// (two K=4 steps cover D=8), then sqrt/exp post-processing on VALU/TRANS.
// dist^2 = n_i + n_j - 2*G_ij + 2*eps*(s_i - s_j) + D*eps^2
// ---------------------------------------------------------------------------
__global__ __launch_bounds__(256) void lsm_pair_kernel(
    const float* __restrict__ ws, float* __restrict__ partial1) {
  __shared__ float red[256];
  const float* Zs = ws + WS_ZS;
  const float* bs = ws + WS_BS;
  const float* ns = ws + WS_NS;
  const float* ss = ws + WS_SS;

  int wave = threadIdx.x >> 5;
  int lane = threadIdx.x & 31;
  int tile = blockIdx.x * 8 + wave;
  int rowBase = (tile >> 8) * 16;   // tile / 256
  int colBase = (tile & 255) * 16;  // tile % 256
  int half = lane >> 4;             // K-half select per ISA layout
  int r = lane & 15;                // M (for A) / N (for B/C/D)

  v8f c = {0.f, 0.f, 0.f, 0.f, 0.f, 0.f, 0.f, 0.f};
#pragma unroll
  for (int k0 = 0; k0 < 8; k0 += 4) {
    v2f a, b;
    int ka = k0 + half * 2;
    a.x = Zs[(rowBase + r) * 8 + ka + 0];
    a.y = Zs[(rowBase + r) * 8 + ka + 1];
    b.x = Zs[(colBase + r) * 8 + ka + 0];
    b.y = Zs[(colBase + r) * 8 + ka + 1];
    // 8-arg f32 WMMA: (neg_a, A, neg_b, B, c_mod, C, reuse_a, reuse_b)
    c = __builtin_amdgcn_wmma_f32_16x16x4_f32(
        false, a, false, b, (short)0, c, false, false);
  }

  int ng = colBase + r;             // column sample index (j)
  float bn = bs[ng], nn = ns[ng], sn = ss[ng];
  float acc = 0.f;
#pragma unroll
  for (int v = 0; v < 8; ++v) {
    int mg = rowBase + half * 8 + v;  // row sample index (i) per C/D layout
    float d2 = ns[mg] + nn - 2.f * c[v] +
               2.f * EPSF * (ss[mg] - sn) + 8.f * EPSF * EPSF;
    float dist = sqrtf(fmaxf(d2, 0.f));
    float val = expf(bs[mg] + bn - dist);
    acc += (mg != ng) ? val : 0.f;    // exclude diagonal (trace)
  }

  red[threadIdx.x] = acc;
  __syncthreads();
#pragma unroll
  for (int s2 = 128; s2 > 0; s2 >>= 1) {
    if (threadIdx.x < s2) red[threadIdx.x] += red[threadIdx.x + s2];
    __syncthreads();
  }
  if (threadIdx.x == 0) partial1[blockIdx.x] = -0.5f * red[0];
}

// ---------------------------------------------------------------------------
// Kernel 3: edge term over E gathered edges (L2-resident Z table).
// ---------------------------------------------------------------------------
__global__ __launch_bounds__(256) void lsm_edge_kernel(
    const float* __restrict__ beta, const float* __restrict__ Z,
    const int* __restrict__ si, const int* __restrict__ sj,
    float* __restrict__ partial2) {
  __shared__ float red[256];
  float acc = 0.f;
  for (int e = blockIdx.x * blockDim.x + threadIdx.x; e < NE;
       e += gridDim.x * blockDim.x) {
    int i = si[e], j = sj[e];
    const float4* zi = (const float4*)(Z + (size_t)i * 8);
    const float4* zj = (const float4*)(Z + (size_t)j * 8);
    float4 a0 = zi[0], a1 = zi[1], b0 = zj[0], b1 = zj[1];
    float d, s = 0.f;
    d = a0.x - b0.x + EPSF; s = fmaf(d, d, s);
    d = a0.y - b0.y + EPSF; s = fmaf(d, d, s);
    d = a0.z - b0.z + EPSF; s = fmaf(d, d, s);
    d = a0.w - b0.w + EPSF; s = fmaf(d, d, s);
    d = a1.x - b1.x + EPSF; s = fmaf(d, d, s);
    d = a1.y - b1.y + EPSF; s = fmaf(d, d, s);
    d = a1.z - b1.z + EPSF; s = fmaf(d, d, s);
    d = a1.w - b1.w + EPSF; s = fmaf(d, d, s);
    acc += beta[i] + beta[j] - sqrtf(s);
  }
  red[threadIdx.x] = acc;
  __syncthreads();
#pragma unroll
  for (int s2 = 128; s2 > 0; s2 >>= 1) {
    if (threadIdx.x < s2) red[threadIdx.x] += red[threadIdx.x + s2];
    __syncthreads();
  }
  if (threadIdx.x == 0) partial2[blockIdx.x] = red[0];
}

// ---------------------------------------------------------------------------
// Kernel 4: deterministic final reduction (fixed-order strided + tree).
// ---------------------------------------------------------------------------
__global__ __launch_bounds__(256) void lsm_final_kernel(
    const float* __restrict__ partial1, const float* __restrict__ partial2,
    float* __restrict__ out) {
  __shared__ float red[256];
  float acc = 0.f;
  for (int i = threadIdx.x; i < PAIR_BLOCKS; i += 256) acc += partial1[i];
  for (int i = threadIdx.x; i < EDGE_BLOCKS; i += 256) acc += partial2[i];
  red[threadIdx.x] = acc;
  __syncthreads();
#pragma unroll
  for (int s2 = 128; s2 > 0; s2 >>= 1) {
    if (threadIdx.x < s2) red[threadIdx.x] += red[threadIdx.x + s2];
    __syncthreads();
  }
  if (threadIdx.x == 0) out[0] = red[0];  // z_pdist2 - z_pdist1
}

// ---------------------------------------------------------------------------
extern "C" void kernel_launch(void* const* d_in, const int* in_sizes, int n_in,
                              void* d_out, int out_size, void* d_ws,
                              size_t ws_size, hipStream_t stream) {
  (void)in_sizes; (void)n_in; (void)out_size; (void)ws_size;
  const float* beta = (const float*)d_in[0];
  const float* Z    = (const float*)d_in[1];
  const int* sidx   = (const int*)d_in[2];
  const int* si     = (const int*)d_in[3];
  const int* sj     = (const int*)d_in[4];
  float* ws  = (float*)d_ws;
  float* out = (float*)d_out;

  lsm_gather_kernel<<<NS / 256, 256, 0, stream>>>(beta, Z, sidx, ws);
  lsm_pair_kernel<<<PAIR_BLOCKS, 256, 0, stream>>>(ws, ws + WS_P1);
  lsm_edge_kernel<<<EDGE_BLOCKS, 256, 0, stream>>>(beta, Z, si, sj, ws + WS_P2);
  lsm_final_kernel<<<1, 256, 0, stream>>>(ws + WS_P1, ws + WS_P2, out);
}